// MultiHeadExtractor_61984968015974
// MI455X (gfx1250) — compile-verified
//
#include <hip/hip_runtime.h>

#define NUM_HEADS       16
#define SUBSET_SIZE     256
#define NODES_PER_GRAPH 4096
#define BATCH_SIZE      128
#define NODE_DIM        128
#define D3              (NODE_DIM * 3)   // 384  (mean|max|sum)
#define D2              (NODE_DIM * 2)   // 256  (hidden dim)
#define HD              128              // head output dim
#define LN_EPS          1e-5f

typedef __attribute__((ext_vector_type(2))) float v2f;
typedef __attribute__((ext_vector_type(8))) float v8f;

// ---------------------------------------------------------------------------
// Kernel 1: irregular gather + segment mean/max/sum -> combined [B*H, 384]
// One block per (graph, head); one lane per embedding dim; indices staged in
// LDS so the 256-row loop issues only coalesced 512B row reads.
// ---------------------------------------------------------------------------
__global__ __launch_bounds__(128) void gather_reduce_kernel(
    const float* __restrict__ emb,      // [B*4096, 128]
    const int*   __restrict__ subset,   // [H*256]
    float*       __restrict__ combined) // [B*H, 384]
{
    __shared__ int sidx[SUBSET_SIZE];
    const int bh = blockIdx.x;              // b*H + h
    const int b  = bh / NUM_HEADS;
    const int h  = bh % NUM_HEADS;
    const int t  = threadIdx.x;             // dim 0..127

    const int* idx = subset + h * SUBSET_SIZE;
    sidx[t]       = idx[t];
    sidx[t + 128] = idx[t + 128];
    __syncthreads();

    const float* base = emb + (size_t)b * NODES_PER_GRAPH * NODE_DIM;
    float s = 0.0f;
    float m = -3.402823466e38f;
#pragma unroll 4
    for (int i = 0; i < SUBSET_SIZE; ++i) {
        float v = base[(size_t)sidx[i] * NODE_DIM + t];
        s += v;
        m = fmaxf(m, v);
    }
    float* o = combined + (size_t)bh * D3;
    o[t]                = s * (1.0f / SUBSET_SIZE); // mean
    o[NODE_DIM + t]     = m;                        // max (counts always 256 > 0)
    o[2 * NODE_DIM + t] = s;                        // sum
}

// ---------------------------------------------------------------------------
// Kernel 2: fused  X@W1 + b1 -> LN -> ReLU -> @W2 + b2 -> LN -> ReLU
// One block per (16-row M tile, head). 16 waves; each wave owns a 16-col
// WMMA N-tile. All intermediates live in LDS.
//
// V_WMMA_F32_16X16X4_F32 layouts (wave32):
//   A 16x4 : lane&15 = M row; lane>>4 selects K pair {0,1}/{2,3} in v[0..1]
//   B 4x16 : lane&15 = N col; lane>>4 selects K pair          in v[0..1]
//   C 16x16: lane&15 = N col; VGPR j = row j + 8*(lane>>4)
// ---------------------------------------------------------------------------
__global__ __launch_bounds__(512) void fused_mlp_kernel(
    const float* __restrict__ X,   // combined [B*H, 384]
    const float* __restrict__ W1,  // [H, 384, 256]
    const float* __restrict__ B1,  // [H, 256]
    const float* __restrict__ G1,  // [256]
    const float* __restrict__ E1,  // [256]
    const float* __restrict__ W2,  // [H, 256, 128]
    const float* __restrict__ B2,  // [H, 128]
    const float* __restrict__ G2,  // [128]
    const float* __restrict__ E2,  // [128]
    float*       __restrict__ out) // [B, H, 128]
{
    __shared__ float xs [16 * D3];  // 24 KB : A tile for proj1
    __shared__ float h1s[16 * D2];  // 16 KB : hidden activations
    __shared__ float o2s[16 * HD];  //  8 KB : pre-LN2 outputs

    const int h    = blockIdx.x % NUM_HEADS;
    const int m0   = (blockIdx.x / NUM_HEADS) * 16;   // first batch row
    const int tid  = threadIdx.x;
    const int w    = tid >> 5;                        // wave id 0..15
    const int lane = tid & 31;

    // ---- stage 16x384 A tile (coalesced: 384 contiguous floats per row) ----
    for (int i = tid; i < 16 * D3; i += 512) {
        int r = i / D3, c = i - r * D3;
        xs[i] = X[((size_t)(m0 + r) * NUM_HEADS + h) * D3 + c];
    }
    __syncthreads();

    const int arow  = lane & 15;          // A row / also B,C column index
    const int kb    = (lane >> 4) << 1;   // K sub-offset: 0 or 2
    const int ccol  = lane & 15;
    const int crow0 = (lane >> 4) << 3;   // C rows 0..7 or 8..15

    // ---------------- proj1: [16x384] @ [384x256], wave w -> cols n0..n0+15
    {
        const int n0 = w * 16;
        v8f c = {0.f, 0.f, 0.f, 0.f, 0.f, 0.f, 0.f, 0.f};
        const float* wp = W1 + ((size_t)h * D3) * D2 + n0 + ccol;
#pragma unroll 4
        for (int k0 = 0; k0 < D3; k0 += 4) {
            v2f a, bm;
            a.x = xs[arow * D3 + k0 + kb];
            a.y = xs[arow * D3 + k0 + kb + 1];
            const float* wk = wp + (size_t)(k0 + kb) * D2;
            bm.x = wk[0];
            bm.y = wk[D2];
            c = __builtin_amdgcn_wmma_f32_16x16x4_f32(
                    false, a, false, bm, (short)0, c, false, false);
        }
#pragma unroll
        for (int j = 0; j < 8; ++j) {
            int r  = crow0 + j;
            int cc = n0 + ccol;
            h1s[r * D2 + cc] = c[j] + B1[h * D2 + cc];
        }
    }
    __syncthreads();

    // ---------------- LayerNorm1 + ReLU (one wave per row, wave32 shuffles)
    {
        const int r = tid >> 5;
        float s = 0.f, ss = 0.f;
#pragma unroll
        for (int j = 0; j < D2 / 32; ++j) {
            float v = h1s[r * D2 + lane + 32 * j];
            s += v; ss += v * v;
        }
#pragma unroll
        for (int o = 16; o > 0; o >>= 1) {
            s  += __shfl_xor(s, o, 32);
            ss += __shfl_xor(ss, o, 32);
        }
        float mu  = s * (1.f / D2);
        float inv = rsqrtf(ss * (1.f / D2) - mu * mu + LN_EPS);
#pragma unroll
        for (int j = 0; j < D2 / 32; ++j) {
            int   cc = lane + 32 * j;
            float v  = h1s[r * D2 + cc];
            v = (v - mu) * inv * G1[cc] + E1[cc];
            h1s[r * D2 + cc] = fmaxf(v, 0.f);
        }
    }
    __syncthreads();

    // ---------------- proj2: [16x256] @ [256x128], waves 0..7 (wave-uniform)
    if (w < 8) {
        const int n0 = w * 16;
        v8f c = {0.f, 0.f, 0.f, 0.f, 0.f, 0.f, 0.f, 0.f};
        const float* wp = W2 + ((size_t)h * D2) * HD + n0 + ccol;
#pragma unroll 4
        for (int k0 = 0; k0 < D2; k0 += 4) {
            v2f a, bm;
            a.x = h1s[arow * D2 + k0 + kb];
            a.y = h1s[arow * D2 + k0 + kb + 1];
            const float* wk = wp + (size_t)(k0 + kb) * HD;
            bm.x = wk[0];
            bm.y = wk[HD];
            c = __builtin_amdgcn_wmma_f32_16x16x4_f32(
                    false, a, false, bm, (short)0, c, false, false);
        }
#pragma unroll
        for (int j = 0; j < 8; ++j) {
            int r  = crow0 + j;
            int cc = n0 + ccol;
            o2s[r * HD + cc] = c[j] + B2[h * HD + cc];
        }
    }
    __syncthreads();

    // ---------------- LayerNorm2 + ReLU + store to [B, H, 128]
    {
        const int r = tid >> 5;
        float s = 0.f, ss = 0.f;
#pragma unroll
        for (int j = 0; j < HD / 32; ++j) {
            float v = o2s[r * HD + lane + 32 * j];
            s += v; ss += v * v;
        }
#pragma unroll
        for (int o = 16; o > 0; o >>= 1) {
            s  += __shfl_xor(s, o, 32);
            ss += __shfl_xor(ss, o, 32);
        }
        float mu  = s * (1.f / HD);
        float inv = rsqrtf(ss * (1.f / HD) - mu * mu + LN_EPS);
#pragma unroll
        for (int j = 0; j < HD / 32; ++j) {
            int   cc = lane + 32 * j;
            float v  = o2s[r * HD + cc];
            v = fmaxf((v - mu) * inv * G2[cc] + E2[cc], 0.f);
            out[((size_t)(m0 + r) * NUM_HEADS + h) * HD + cc] = v;
        }
    }
}

// ---------------------------------------------------------------------------
extern "C" void kernel_launch(void* const* d_in, const int* in_sizes, int n_in,
                              void* d_out, int out_size, void* d_ws, size_t ws_size,
                              hipStream_t stream) {
    (void)in_sizes; (void)n_in; (void)out_size; (void)ws_size;
    const float* emb    = (const float*)d_in[0];
    const float* W1     = (const float*)d_in[1];
    const float* B1     = (const float*)d_in[2];
    const float* G1     = (const float*)d_in[3];
    const float* E1     = (const float*)d_in[4];
    const float* W2     = (const float*)d_in[5];
    const float* B2     = (const float*)d_in[6];
    const float* G2     = (const float*)d_in[7];
    const float* E2     = (const float*)d_in[8];
    // d_in[9]  = batch (contiguous b*4096 layout -> offsets computed directly)
    const int*   subset = (const int*)d_in[10];
    // d_in[11] = head_batch (head h owns subset rows [h*256, (h+1)*256))
    // d_in[12] = batch_size scalar (compile-time constant here)

    float* combined = (float*)d_ws;  // [B*H, 384] = 3 MB

    gather_reduce_kernel<<<BATCH_SIZE * NUM_HEADS, 128, 0, stream>>>(
        emb, subset, combined);

    fused_mlp_kernel<<<(BATCH_SIZE / 16) * NUM_HEADS, 512, 0, stream>>>(
        combined, W1, B1, G1, E1, W2, B2, G2, E2, (float*)d_out);
}